// torch_r_dyad_79800492360004
// MI455X (gfx1250) — compile-verified
//
#include <hip/hip_runtime.h>
#include <math.h>
#include <stdint.h>

// Per-ray polarization dyad: out = m_r @ diag(fresnel) @ m_i  (complex64 2x2 per ray).
// Memory-bound elementwise kernel; CDNA5 path = async global->LDS DMA staging.
// Hot kernel handles only full 1024-ray blocks (branch-free: pure async DMA + ds_load).
// A separate tail kernel (direct global loads) covers N % 1024 leftovers.

namespace {

constexpr double kF     = 3.6e9;
constexpr double kE0    = 8.8541878188e-12;
constexpr double kTwoPi = 6.283185307179586;

struct V3 { float x, y, z; };

__device__ __forceinline__ V3 cross3(V3 a, V3 b) {
  return { a.y*b.z - a.z*b.y, a.z*b.x - a.x*b.z, a.x*b.y - a.y*b.x };
}
__device__ __forceinline__ float dot3(V3 a, V3 b) {
  return a.x*b.x + a.y*b.y + a.z*b.z;
}
// Spherical unit vectors without trig: for unit d,
// vt = [cosT*cosP, cosT*sinP, -sinT], vp = [-sinP, cosP, 0],
// with sinT = sqrt(x^2+y^2), cosT = z, cosP = x/sinT, sinP = y/sinT.
__device__ __forceinline__ void sph_basis(V3 d, V3& vt, V3& vp) {
  float s2  = d.x*d.x + d.y*d.y;
  float inv = rsqrtf(s2);
  vt = { d.z*d.x*inv, d.z*d.y*inv, -s2*inv };
  vp = { -d.y*inv,    d.x*inv,      0.0f   };
}

// Full per-ray math shared by both kernels.
__device__ __forceinline__ void compute_ray(V3 inc, V3 dep, V3 nor,
                                            float er_re, float er_im,
                                            float4& o0, float4& o1) {
  // ---- local polarization bases ----
  V3 c1 = cross3(nor, inc);
  V3 u1 = cross3(inc, c1);                       // e_pai (unnormalized)
  float inv1 = rsqrtf(dot3(u1, u1));
  V3 e_pai = { u1.x*inv1, u1.y*inv1, u1.z*inv1 };
  float ndi = dot3(nor, inc);
  V3 t = cross3(nor, e_pai);
  float invn = 1.0f / ndi;
  V3 e_pei = { t.x*invn, t.y*invn, t.z*invn };   // == e_per
  V3 c2 = cross3(nor, dep);
  V3 u2 = cross3(dep, c2);
  float inv2 = rsqrtf(dot3(u2, u2));
  V3 e_par = { u2.x*inv2, u2.y*inv2, u2.z*inv2 };

  V3 vti, vpi, vtr, vpr;
  sph_basis(inc, vti, vpi);
  sph_basis(dep, vtr, vpr);

  // ---- basis-change matrices ----
  float mi00 = dot3(e_pai, vti), mi01 = dot3(e_pai, vpi);
  float mi10 = dot3(e_pei, vti), mi11 = dot3(e_pei, vpi);
  float mr00 = dot3(vtr, e_par), mr01 = dot3(vtr, e_pei);
  float mr10 = dot3(vpr, e_par), mr11 = dot3(vpr, e_pei);

  // ---- Fresnel coefficients (alpha = -arccos(ndi): ca = ndi, sin^2 = 1-ndi^2) ----
  float ca = ndi;
  float a  = er_re - (1.0f - ndi*ndi);           // Re(er - sin^2 a)
  float b  = er_im;                              // Im(er)
  float m  = sqrtf(a*a + b*b);
  float sq_re = sqrtf(0.5f * fmaxf(m + a, 0.0f));
  float sq_im = copysignf(sqrtf(0.5f * fmaxf(m - a, 0.0f)), b);

  // r00 = (er*ca - sq)/(er*ca + sq)
  float n0r = er_re*ca - sq_re, n0i = er_im*ca - sq_im;
  float d0r = er_re*ca + sq_re, d0i = er_im*ca + sq_im;
  float id0 = 1.0f / (d0r*d0r + d0i*d0i);
  float r0re = (n0r*d0r + n0i*d0i) * id0;
  float r0im = (n0i*d0r - n0r*d0i) * id0;
  // r11 = (ca - sq)/(ca + sq)
  float n1r = ca - sq_re, n1i = -sq_im;
  float d1r = ca + sq_re, d1i =  sq_im;
  float id1 = 1.0f / (d1r*d1r + d1i*d1i);
  float r1re = (n1r*d1r + n1i*d1i) * id1;
  float r1im = (n1i*d1r - n1r*d1i) * id1;

  // ---- out[i][k] = mr[i][0]*r0*mi[0][k] + mr[i][1]*r1*mi[1][k] ----
  float w000 = mr00*mi00, w001 = mr00*mi01;
  float w010 = mr10*mi00, w011 = mr10*mi01;
  float w100 = mr01*mi10, w101 = mr01*mi11;
  float w110 = mr11*mi10, w111 = mr11*mi11;

  o0 = { w000*r0re + w100*r1re, w000*r0im + w100*r1im,
         w001*r0re + w101*r1re, w001*r0im + w101*r1im };
  o1 = { w010*r0re + w110*r1re, w010*r0im + w110*r1im,
         w011*r0re + w111*r1re, w011*r0im + w111*r1im };
}

} // namespace

#define RAYS_PER_BLOCK 1024
#define RAYS_PER_WAVE  128
#define FLOATS_PER_WAVE (3 * RAYS_PER_WAVE * 3)   // 1152 floats = 4608 B per wave

// ---------------- Hot kernel: full blocks only, branch-free ----------------
__global__ __launch_bounds__(256) void dyad_fresnel_main(
    const float* __restrict__ x,     // [3, N, 3]: inc, dep, nor
    const float* __restrict__ eps,   // [1]
    const float* __restrict__ cond,  // [1]
    float* __restrict__ out,         // [N, 2, 2] complex64 -> N*8 floats
    int N)                            // only used for region strides
{
  __shared__ __align__(16) float smem[8 * FLOATS_PER_WAVE]; // 36864 B

  const int tid  = threadIdx.x;
  const int wave = tid >> 5;
  const int lane = tid & 31;
  const int wb   = wave * FLOATS_PER_WAVE;
  const size_t waveBase =
      (size_t)blockIdx.x * RAYS_PER_BLOCK + (size_t)wave * RAYS_PER_WAVE;

  // Stage this wave's 128 rays for all 3 regions: 3 x 1536 B,
  // 9 aligned b128 async DMA transfers per lane, fully coalesced.
  {
    const uint32_t lds0 = (uint32_t)(uintptr_t)(void*)&smem[wb];
    #pragma unroll
    for (int r = 0; r < 3; ++r) {
      const float* gb = x + (size_t)r * (size_t)N * 3 + waveBase * 3;
      #pragma unroll
      for (int c = 0; c < 3; ++c) {
        uint32_t loff  = lds0 + (uint32_t)(r * 1536 + c * 512 + lane * 16);
        const float* g = gb + (c * 128 + lane * 4);
        asm volatile("global_load_async_to_lds_b128 %0, %1, off"
                     :: "v"(loff), "v"(g) : "memory");
      }
    }
#if __has_builtin(__builtin_amdgcn_s_wait_asynccnt)
    __builtin_amdgcn_s_wait_asynccnt(0);
    asm volatile("" ::: "memory");
#else
    asm volatile("s_wait_asynccnt 0" ::: "memory");
#endif
  }

  const float er_re = eps[0];
  const float er_im = -cond[0] * (float)(1.0 / (kTwoPi * kF * kE0));

  #pragma unroll
  for (int k = 0; k < 4; ++k) {
    const int l = lane + 32 * k;      // bank-conflict-free: stride 3 words
    const size_t ray = waveBase + l;

    V3 inc = { smem[wb +       l*3 + 0], smem[wb +       l*3 + 1], smem[wb +       l*3 + 2] };
    V3 dep = { smem[wb + 384 + l*3 + 0], smem[wb + 384 + l*3 + 1], smem[wb + 384 + l*3 + 2] };
    V3 nor = { smem[wb + 768 + l*3 + 0], smem[wb + 768 + l*3 + 1], smem[wb + 768 + l*3 + 2] };

    float4 o0, o1;
    compute_ray(inc, dep, nor, er_re, er_im, o0, o1);

    float4* po = reinterpret_cast<float4*>(out + ray * 8);
    po[0] = o0;   // 32 B/ray, contiguous per k-step -> coalesced b128 stores
    po[1] = o1;
  }
}

// ---------------- Tail kernel: direct global loads, guarded ----------------
__global__ __launch_bounds__(256) void dyad_fresnel_tail(
    const float* __restrict__ x,
    const float* __restrict__ eps,
    const float* __restrict__ cond,
    float* __restrict__ out,
    int N, int tailStart)
{
  const int i = tailStart + blockIdx.x * blockDim.x + threadIdx.x;
  if (i >= N) return;
  const size_t ray = (size_t)i;

  const float* p0 = x + ray * 3;
  const float* p1 = x + (size_t)N * 3 + ray * 3;
  const float* p2 = x + (size_t)N * 6 + ray * 3;
  V3 inc = { p0[0], p0[1], p0[2] };
  V3 dep = { p1[0], p1[1], p1[2] };
  V3 nor = { p2[0], p2[1], p2[2] };

  const float er_re = eps[0];
  const float er_im = -cond[0] * (float)(1.0 / (kTwoPi * kF * kE0));

  float4 o0, o1;
  compute_ray(inc, dep, nor, er_re, er_im, o0, o1);

  float4* po = reinterpret_cast<float4*>(out + ray * 8);
  po[0] = o0;
  po[1] = o1;
}

extern "C" void kernel_launch(void* const* d_in, const int* in_sizes, int n_in,
                              void* d_out, int out_size, void* d_ws, size_t ws_size,
                              hipStream_t stream) {
  const float* x    = (const float*)d_in[0];   // [3, N, 3] float32
  const float* eps  = (const float*)d_in[1];   // [1] float32
  const float* cond = (const float*)d_in[2];   // [1] float32
  const int N = in_sizes[0] / 9;

  const int fullBlocks = N / RAYS_PER_BLOCK;           // staged, branch-free
  const int tailStart  = fullBlocks * RAYS_PER_BLOCK;
  const int tailCount  = N - tailStart;

  if (fullBlocks > 0) {
    dyad_fresnel_main<<<fullBlocks, 256, 0, stream>>>(x, eps, cond, (float*)d_out, N);
  }
  if (tailCount > 0) {
    const int tb = (tailCount + 255) / 256;
    dyad_fresnel_tail<<<tb, 256, 0, stream>>>(x, eps, cond, (float*)d_out, N, tailStart);
  }
}